// LocalCrossAttention_10892037063093
// MI455X (gfx1250) — compile-verified
//
#include <hip/hip_runtime.h>
#include <math.h>

#define FEATURE_DIM 256
#define NUM_HEADS   8
#define HEAD_DIM    32
#define N_CUR       4096
#define N_HIST      4096
#define RADIUS2     9.0f

typedef __attribute__((ext_vector_type(2))) float v2f;
typedef __attribute__((ext_vector_type(8))) float v8f;

static __device__ __forceinline__ v8f wmma4(v2f a, v2f b, v8f c) {
    // 8 args: (neg_a, A, neg_b, B, c_mod, C, reuse_a, reuse_b)
    return __builtin_amdgcn_wmma_f32_16x16x4_f32(false, a, false, b, (short)0, c,
                                                 false, false);
}

// C[M x 256] = A[M x 256] @ W^T + bias   (W is [256 x 256] row-major, reference's nn.Linear)
// One wave per 16x16 output tile.
__global__ void __launch_bounds__(256)
proj16_kernel(const float* __restrict__ A, const float* __restrict__ W,
              const float* __restrict__ bias, float* __restrict__ C, int M) {
    const int wave = (blockIdx.x * blockDim.x + threadIdx.x) >> 5;
    const int lane = threadIdx.x & 31;
    const int ntiles_n = FEATURE_DIM / 16;            // 16
    const int mt = wave / ntiles_n;
    const int nt = wave % ntiles_n;
    if (mt * 16 >= M) return;

    const int half = lane >> 4;                        // K-half selector
    const int l    = lane & 15;
    const int col  = nt * 16 + l;

    const float bcol = bias[col];
    v8f acc = {bcol, bcol, bcol, bcol, bcol, bcol, bcol, bcol};

    const float* arow = A + (size_t)(mt * 16 + l) * FEATURE_DIM;  // A fragment row
    const float* brow = W + (size_t)col * FEATURE_DIM;            // W^T: K-contiguous

#pragma unroll 8
    for (int k = 0; k < FEATURE_DIM; k += 4) {
        const int ka = k + 2 * half;
        v2f a = *(const v2f*)(arow + ka);
        v2f b = *(const v2f*)(brow + ka);
        acc = wmma4(a, b, acc);
    }

#pragma unroll
    for (int i = 0; i < 8; ++i) {
        const int row = mt * 16 + i + 8 * half;        // C layout: M = vgpr + 8*half
        C[(size_t)row * FEATURE_DIM + col] = acc[i];
    }
}

// Flash attention with distance mask. One wave (32 threads) per (query-tile, head).
__global__ void __launch_bounds__(32)
attn_kernel(const float* __restrict__ Q, const float* __restrict__ K,
            const float* __restrict__ V,
            const float* __restrict__ cur_xyz, const float* __restrict__ hist_xyz,
            float* __restrict__ AO) {
    __shared__ float Pbuf[16][17];                     // padded: C-layout -> A-layout bounce

    const int qt   = blockIdx.x / NUM_HEADS;
    const int h    = blockIdx.x % NUM_HEADS;
    const int lane = threadIdx.x & 31;
    const int half = lane >> 4;
    const int l    = lane & 15;
    const int hb   = h * HEAD_DIM;
    const float scale = 0.17677669529663687f;          // 1/sqrt(32)

    // Q A-fragments for the 8 K-chunks of head_dim, pre-scaled by 1/sqrt(d)
    v2f qfrag[8];
    {
        const float* qrow = Q + (size_t)(qt * 16 + l) * FEATURE_DIM + hb;
#pragma unroll
        for (int kk = 0; kk < 8; ++kk) {
            v2f t = *(const v2f*)(qrow + 4 * kk + 2 * half);
            qfrag[kk] = t * scale;
        }
    }

    // query coords for this lane's 8 C-fragment rows
    float qx[8], qy[8], qz[8];
#pragma unroll
    for (int i = 0; i < 8; ++i) {
        const int r = qt * 16 + i + 8 * half;
        qx[i] = cur_xyz[3 * r + 0];
        qy[i] = cur_xyz[3 * r + 1];
        qz[i] = cur_xyz[3 * r + 2];
    }

    v8f o0 = {0.f, 0.f, 0.f, 0.f, 0.f, 0.f, 0.f, 0.f};
    v8f o1 = o0;
    float mrow[8], lrow[8];
#pragma unroll
    for (int i = 0; i < 8; ++i) { mrow[i] = -1e30f; lrow[i] = 0.f; }

    for (int kt = 0; kt < N_HIST / 16; ++kt) {
        const int kbase = kt * 16;

        // ---- S = Q @ K^T (16x16), 8 WMMAs over head_dim ----
        v8f s = {0.f, 0.f, 0.f, 0.f, 0.f, 0.f, 0.f, 0.f};
        const float* krow = K + (size_t)(kbase + l) * FEATURE_DIM + hb;
#pragma unroll
        for (int kk = 0; kk < 8; ++kk) {
            v2f b = *(const v2f*)(krow + 4 * kk + 2 * half);
            s = wmma4(qfrag[kk], b, s);
        }

        // ---- locality mask on the C fragment (lane owns key column l) ----
        {
            const int key = kbase + l;
            const float kx = hist_xyz[3 * key + 0];
            const float ky = hist_xyz[3 * key + 1];
            const float kz = hist_xyz[3 * key + 2];
#pragma unroll
            for (int i = 0; i < 8; ++i) {
                const float dx = qx[i] - kx, dy = qy[i] - ky, dz = qz[i] - kz;
                if (dx * dx + dy * dy + dz * dz > RADIUS2) s[i] = -1e30f;
            }
        }

        // ---- online softmax: per-VGPR row stats, 16-lane butterfly reductions ----
#pragma unroll
        for (int i = 0; i < 8; ++i) {
            float v = s[i];
            v = fmaxf(v, __shfl_xor(v, 1, 32));
            v = fmaxf(v, __shfl_xor(v, 2, 32));
            v = fmaxf(v, __shfl_xor(v, 4, 32));
            v = fmaxf(v, __shfl_xor(v, 8, 32));
            const float mnew = fmaxf(mrow[i], v);
            const float sc   = __expf(mrow[i] - mnew);   // ==0 when leaving all-masked prefix
            mrow[i] = mnew;
            float p = __expf(s[i] - mnew);
            s[i] = p;
            float rs = p;
            rs += __shfl_xor(rs, 1, 32);
            rs += __shfl_xor(rs, 2, 32);
            rs += __shfl_xor(rs, 4, 32);
            rs += __shfl_xor(rs, 8, 32);
            lrow[i] = lrow[i] * sc + rs;
            o0[i] *= sc;
            o1[i] *= sc;
        }

        // ---- P: C-layout -> LDS -> A-layout fragments ----
        __syncthreads();
#pragma unroll
        for (int i = 0; i < 8; ++i) Pbuf[i + 8 * half][l] = s[i];
        __syncthreads();

        // ---- O += P @ V (16x32), two N-tiles, 4 K-chunks each ----
        const float* vbase = V + (size_t)kbase * FEATURE_DIM + hb;
#pragma unroll
        for (int kk = 0; kk < 4; ++kk) {
            const int kr = 4 * kk + 2 * half;
            v2f a;
            a.x = Pbuf[l][kr + 0];
            a.y = Pbuf[l][kr + 1];
            v2f b0, b1;
            b0.x = vbase[(size_t)(kr + 0) * FEATURE_DIM + l];
            b0.y = vbase[(size_t)(kr + 1) * FEATURE_DIM + l];
            b1.x = vbase[(size_t)(kr + 0) * FEATURE_DIM + 16 + l];
            b1.y = vbase[(size_t)(kr + 1) * FEATURE_DIM + 16 + l];
            o0 = wmma4(a, b0, o0);
            o1 = wmma4(a, b1, o1);
        }
        __syncthreads();
    }

    // ---- finalize: divide by row sum, scatter to AO[q][h*32 + d] ----
#pragma unroll
    for (int i = 0; i < 8; ++i) {
        const float inv = 1.0f / lrow[i];
        const int row = qt * 16 + i + 8 * half;
        AO[(size_t)row * FEATURE_DIM + hb + l]      = o0[i] * inv;
        AO[(size_t)row * FEATURE_DIM + hb + 16 + l] = o1[i] * inv;
    }
}

extern "C" void kernel_launch(void* const* d_in, const int* in_sizes, int n_in,
                              void* d_out, int out_size, void* d_ws, size_t ws_size,
                              hipStream_t stream) {
    const float* cur_feats   = (const float*)d_in[0];
    const float* hist_feats  = (const float*)d_in[1];
    const float* cur_coords  = (const float*)d_in[2];
    const float* hist_coords = (const float*)d_in[3];
    const float* Wq = (const float*)d_in[4];
    const float* bq = (const float*)d_in[5];
    const float* Wk = (const float*)d_in[6];
    const float* bk = (const float*)d_in[7];
    const float* Wv = (const float*)d_in[8];
    const float* bv = (const float*)d_in[9];
    const float* Wo = (const float*)d_in[10];
    const float* bo = (const float*)d_in[11];
    float* out = (float*)d_out;

    float* Q  = (float*)d_ws;
    float* K  = Q  + (size_t)N_HIST * FEATURE_DIM;
    float* V  = K  + (size_t)N_HIST * FEATURE_DIM;
    float* AO = V  + (size_t)N_HIST * FEATURE_DIM;

    // projection grids: one wave per 16x16 tile -> (M/16)*(256/16) waves
    const int waves = (N_CUR / 16) * (FEATURE_DIM / 16);   // 4096
    const dim3 pblk(256);
    const dim3 pgrd(waves * 32 / 256);                      // 512

    proj16_kernel<<<pgrd, pblk, 0, stream>>>(cur_feats,  Wq, bq, Q,  N_CUR);
    proj16_kernel<<<pgrd, pblk, 0, stream>>>(hist_feats, Wk, bk, K,  N_HIST);
    proj16_kernel<<<pgrd, pblk, 0, stream>>>(hist_feats, Wv, bv, V,  N_HIST);

    attn_kernel<<<dim3((N_CUR / 16) * NUM_HEADS), dim3(32), 0, stream>>>(
        Q, K, V, cur_coords, hist_coords, AO);

    proj16_kernel<<<pgrd, pblk, 0, stream>>>(AO, Wo, bo, out, N_CUR);
}